// DistMaps_88192858456273
// MI455X (gfx1250) — compile-verified
//
#include <hip/hip_runtime.h>
#include <hip/hip_bf16.h>

typedef __attribute__((ext_vector_type(2))) float v2f;
typedef __attribute__((ext_vector_type(8))) float v8f;

#define NPTS   24          // points per (batch, polarity) group
#define HH     512
#define WW     512
#define NGRP   32          // B*2
#define INV_R  0.2f        // 1 / (NORM_RADIUS * SPATIAL_SCALE)
#define FILLV  1.0e6f      // INVALID_FILL

// d2(point m, pixel n) via one K=4 f32 WMMA:
//   A[m] = [-2*pr', -2*pc', pr'^2+pc'^2, 1]   (invalid/pad: [0,0,1e6,0])
//   B[n] = [ r',     c',    1,           r'^2+c'^2 ]
//   dot  = (r'-pr')^2 + (c'-pc')^2  (or exactly 1e6 for invalid rows)
__global__ __launch_bounds__(128) void distmaps_wmma_kernel(
    const float* __restrict__ coords, float* __restrict__ out)
{
  const int  lane = threadIdx.x & 31;
  const int  wave = threadIdx.x >> 5;     // 4 waves / block
  const int  mrow = lane & 15;
  const bool hi   = lane >= 16;

  const int g        = blockIdx.x >> 5;   // group = b*2 + pol
  const int rowSlice = blockIdx.x & 31;   // 16-row slice of the image

  const int b   = g >> 1;
  const int pol = g & 1;
  const float* cb = coords + ((size_t)(b * 48 + pol * NPTS)) * 3;

  // ---- Build A tiles (points 0..15 and 16..31; 24..31 are padding) ----
  // f32 A 16x4 layout: lanes 0-15 hold K={0,1} in v.{x,y}, lanes 16-31 hold K={2,3}.
  v2f a0, a1;
#pragma unroll
  for (int t = 0; t < 2; ++t) {
    const int p = t * 16 + mrow;
    float pr = -1.0f, pc = -1.0f;
    if (p < NPTS) { pr = cb[p * 3 + 0]; pc = cb[p * 3 + 1]; }
    const bool inval = fmaxf(pr, pc) < 0.0f;   // reference: max(pts) < 0
    const float prs = pr * INV_R;
    const float pcs = pc * INV_R;
    v2f a;
    a.x = hi ? (inval ? FILLV : prs * prs + pcs * pcs)
             : (inval ? 0.0f  : -2.0f * prs);
    a.y = hi ? (inval ? 0.0f  : 1.0f)
             : (inval ? 0.0f  : -2.0f * pcs);
    if (t == 0) a0 = a; else a1 = a;
  }

#pragma unroll 1
  for (int rr = 0; rr < 4; ++rr) {
    const int   h   = rowSlice * 16 + wave * 4 + rr;
    const float rs  = (float)h * INV_R;
    const float rs2 = rs * rs;
    float* __restrict__ orow = out + ((size_t)g * HH + h) * (size_t)WW;

#pragma unroll 1
    for (int w0 = 0; w0 < WW; w0 += 32) {   // two 16-pixel tiles per iter
      float res0 = 0.0f, res1 = 0.0f;
#pragma unroll
      for (int t = 0; t < 2; ++t) {
        // f32 B 4x16 layout: lanes 0-15 hold K={0,1}, lanes 16-31 hold K={2,3}.
        const float cs = (float)(w0 + t * 16 + mrow) * INV_R;
        v2f bm;
        bm.x = hi ? 1.0f            : rs;
        bm.y = hi ? rs2 + cs * cs   : cs;

        v8f cz = {};
        // 8 args: (neg_a, A, neg_b, B, c_mod, C, reuse_a, reuse_b)
        v8f d0 = __builtin_amdgcn_wmma_f32_16x16x4_f32(
            false, a0, false, bm, (short)0, cz, false, false);
        v8f d1 = __builtin_amdgcn_wmma_f32_16x16x4_f32(
            false, a1, false, bm, (short)0, cz, false, false);

        // min over the 16 point-rows held in this lane half...
        float mn = fminf(d0[0], d1[0]);
#pragma unroll
        for (int k = 1; k < 8; ++k) mn = fminf(mn, fminf(d0[k], d1[k]));
        // ...then merge lane halves (points 0-7/16-23 vs 8-15/24-31)
        mn = fminf(mn, __shfl_xor(mn, 16, 32));
        if (t == 0) res0 = mn; else res1 = mn;
      }

      // lanes 0-15 -> tile0 pixels, lanes 16-31 -> tile1 pixels
      const float d2 = fmaxf(hi ? res1 : res0, 0.0f);  // guard tiny negative (cancellation)
      const float xx = 4.0f * __builtin_amdgcn_sqrtf(d2);        // 2 * (2*sqrt(d2))
      const float e  = __builtin_amdgcn_exp2f(xx * 1.4426950408889634f);
      const float v  = 1.0f - 2.0f * __builtin_amdgcn_rcpf(e + 1.0f);  // tanh, sat -> 1
      orow[w0 + lane] = v;
    }
  }
}

extern "C" void kernel_launch(void* const* d_in, const int* in_sizes, int n_in,
                              void* d_out, int out_size, void* d_ws, size_t ws_size,
                              hipStream_t stream) {
  (void)in_sizes; (void)n_in; (void)d_ws; (void)ws_size; (void)out_size;
  // inputs: d_in[0] = x (shape-only, unused), d_in[1] = coords [B,48,3] f32
  const float* coords = (const float*)d_in[1];
  float* out = (float*)d_out;

  dim3 grid(NGRP * (HH / 16));   // 32 groups x 32 row-slices = 1024 blocks
  dim3 block(128);               // 4 wave32
  distmaps_wmma_kernel<<<grid, block, 0, stream>>>(coords, out);
}